// GCNSage_48868137893890
// MI455X (gfx1250) — compile-verified
//
#include <hip/hip_runtime.h>

typedef __attribute__((ext_vector_type(2))) float v2f;
typedef __attribute__((ext_vector_type(8))) float v8f;

#define NNODE 8192
#define DIM   128
#define KNBR  8
#define NEGINF -1e30f

__device__ __forceinline__ float4 ld4(const float* p) {
  return *reinterpret_cast<const float4*>(p);
}
__device__ __forceinline__ void st4(float* p, float4 v) {
  *reinterpret_cast<float4*>(p) = v;
}

// ---------------------------------------------------------------------------
// Per-node degree (with dedupe + self-exclusion) -> dinv, keep-mask
// ---------------------------------------------------------------------------
__global__ void deg_kernel(const int* __restrict__ neigh,
                           const int* __restrict__ nbrc,
                           float* __restrict__ dinv,
                           unsigned* __restrict__ keep) {
  int i = blockIdx.x * blockDim.x + threadIdx.x;
  if (i >= NNODE) return;
  int cnt = nbrc[i]; cnt = cnt > KNBR ? KNBR : cnt;
  const int* nb = neigh + i * KNBR;
  unsigned km = 0;
  int deg = 1;  // self loop
  for (int k = 0; k < cnt; ++k) {
    int j = nb[k];
    if (j == i) continue;              // diag already 1.0
    bool dup = false;
    for (int t = 0; t < k; ++t) dup |= (nb[t] == j);
    if (!dup) { km |= (1u << k); ++deg; }
  }
  dinv[i] = rsqrtf((float)deg);
  keep[i] = km;
}

// ---------------------------------------------------------------------------
// Y[m,n] = sum_k A[m,k] * W[n,k] + bias[n]   (optional ReLU)
// One wave per 16x16 tile; V_WMMA_F32_16X16X4_F32 over K in steps of 4.
// ---------------------------------------------------------------------------
__global__ void wmma_gemm_bias(const float* __restrict__ A, int lda,
                               const float* __restrict__ W, int ldw,
                               const float* __restrict__ bias,
                               float* __restrict__ Y, int ldy,
                               int K, int do_relu) {
  const int wave = threadIdx.x >> 5;
  const int lane = threadIdx.x & 31;
  const int half = lane >> 4;          // lanes 16-31 hold K+2,K+3
  const int l16  = lane & 15;
  const int m0 = blockIdx.x * 16;
  const int n0 = (blockIdx.y * (blockDim.x >> 5) + wave) * 16;

  const float* __restrict__ arow = A + (size_t)(m0 + l16) * lda + 2 * half;
  const float* __restrict__ wrow = W + (size_t)(n0 + l16) * ldw + 2 * half;

  v8f c = {};
  for (int k0 = 0; k0 < K; k0 += 4) {
    v2f a = *reinterpret_cast<const v2f*>(arow + k0);
    v2f b = *reinterpret_cast<const v2f*>(wrow + k0);
    // (neg_a, A, neg_b, B, c_mod, C, reuse_a, reuse_b)
    c = __builtin_amdgcn_wmma_f32_16x16x4_f32(false, a, false, b,
                                              (short)0, c, false, false);
  }

  const float bv = bias[n0 + l16];
#pragma unroll
  for (int r = 0; r < 8; ++r) {
    float v = c[r] + bv;
    if (do_relu) v = fmaxf(v, 0.0f);
    Y[(size_t)(m0 + r + 8 * half) * ldy + n0 + l16] = v;
  }
}

// ---------------------------------------------------------------------------
// Wave-local LayerNorm over 128 features (4 per lane), write to dstrow.
// ---------------------------------------------------------------------------
__device__ __forceinline__ void wave_ln_store(float4 v,
                                              const float* __restrict__ g,
                                              const float* __restrict__ b,
                                              float* __restrict__ dstrow,
                                              int lane) {
  float s = v.x + v.y + v.z + v.w;
#pragma unroll
  for (int off = 16; off > 0; off >>= 1) s += __shfl_xor(s, off, 32);
  const float mu = s * (1.0f / (float)DIM);
  const float dx = v.x - mu, dy = v.y - mu, dz = v.z - mu, dw = v.w - mu;
  float q = dx * dx + dy * dy + dz * dz + dw * dw;
#pragma unroll
  for (int off = 16; off > 0; off >>= 1) q += __shfl_xor(q, off, 32);
  const float rinv = rsqrtf(q * (1.0f / (float)DIM) + 1e-5f);
  const int d = lane * 4;
  const float4 gg = ld4(g + d);
  const float4 bb = ld4(b + d);
  float4 o;
  o.x = dx * rinv * gg.x + bb.x;
  o.y = dy * rinv * gg.y + bb.y;
  o.z = dz * rinv * gg.z + bb.z;
  o.w = dw * rinv * gg.w + bb.w;
  st4(dstrow + d, o);
}

// ---------------------------------------------------------------------------
// GCN: h = LN(relu(Ah @ nxt) + resid).  One wave per node.
// ---------------------------------------------------------------------------
__global__ void gcn_prop_ln(const float* __restrict__ nxt,
                            const float* __restrict__ resid,
                            const int* __restrict__ neigh,
                            const float* __restrict__ dinv,
                            const unsigned* __restrict__ keep,
                            const float* __restrict__ g,
                            const float* __restrict__ b,
                            float* __restrict__ dst, int ldd) {
  const int i = (int)((blockIdx.x * blockDim.x + threadIdx.x) >> 5);
  if (i >= NNODE) return;
  const int lane = threadIdx.x & 31;
  const int d = lane * 4;

  const float di = dinv[i];
  float4 t = ld4(nxt + (size_t)i * DIM + d);
  float4 acc = {di * t.x, di * t.y, di * t.z, di * t.w};  // self term

  const unsigned km = keep[i];
  const int* nb = neigh + i * KNBR;
  for (int k = 0; k < KNBR; ++k) {
    if ((km >> k) & 1u) {
      const int j = nb[k];
      const float dj = dinv[j];
      float4 u = ld4(nxt + (size_t)j * DIM + d);
      acc.x += dj * u.x; acc.y += dj * u.y;
      acc.z += dj * u.z; acc.w += dj * u.w;
    }
  }
  acc.x = fmaxf(acc.x * di, 0.0f);
  acc.y = fmaxf(acc.y * di, 0.0f);
  acc.z = fmaxf(acc.z * di, 0.0f);
  acc.w = fmaxf(acc.w * di, 0.0f);

  const float4 r = ld4(resid + (size_t)i * DIM + d);
  float4 v = {acc.x + r.x, acc.y + r.y, acc.z + r.z, acc.w + r.w};
  wave_ln_store(v, g, b, dst + (size_t)i * ldd, lane);
}

// ---------------------------------------------------------------------------
// SAGE: cat[i] = [ selfsrc[i] | maxpool_k poolsrc[neigh[i,k]] ] (leaf -> 0).
// ---------------------------------------------------------------------------
__global__ void sage_pool_cat(const float* __restrict__ selfsrc,
                              const float* __restrict__ poolsrc,
                              const int* __restrict__ neigh,
                              const int* __restrict__ nbrc,
                              float* __restrict__ cat) {
  const int i = (int)((blockIdx.x * blockDim.x + threadIdx.x) >> 5);
  if (i >= NNODE) return;
  const int lane = threadIdx.x & 31;
  const int d = lane * 4;

  st4(cat + (size_t)i * (2 * DIM) + d, ld4(selfsrc + (size_t)i * DIM + d));

  int cnt = nbrc[i]; cnt = cnt > KNBR ? KNBR : cnt;
  float4 m = {NEGINF, NEGINF, NEGINF, NEGINF};
  const int* nb = neigh + i * KNBR;
  for (int k = 0; k < cnt; ++k) {
    const int j = nb[k];
    float4 u = ld4(poolsrc + (size_t)j * DIM + d);
    m.x = fmaxf(m.x, u.x); m.y = fmaxf(m.y, u.y);
    m.z = fmaxf(m.z, u.z); m.w = fmaxf(m.w, u.w);
  }
  if (cnt == 0) m = make_float4(0.f, 0.f, 0.f, 0.f);
  st4(cat + (size_t)i * (2 * DIM) + DIM + d, m);
}

// ---------------------------------------------------------------------------
// s = LN(a + res).  One wave per node.
// ---------------------------------------------------------------------------
__global__ void resid_ln(const float* __restrict__ a,
                         const float* __restrict__ res,
                         const float* __restrict__ g,
                         const float* __restrict__ b,
                         float* __restrict__ dst, int ldd) {
  const int i = (int)((blockIdx.x * blockDim.x + threadIdx.x) >> 5);
  if (i >= NNODE) return;
  const int lane = threadIdx.x & 31;
  const int d = lane * 4;
  const float4 av = ld4(a + (size_t)i * DIM + d);
  const float4 rv = ld4(res + (size_t)i * DIM + d);
  float4 v = {av.x + rv.x, av.y + rv.y, av.z + rv.z, av.w + rv.w};
  wave_ln_store(v, g, b, dst + (size_t)i * ldd, lane);
}

// ---------------------------------------------------------------------------
extern "C" void kernel_launch(void* const* d_in, const int* in_sizes, int n_in,
                              void* d_out, int out_size, void* d_ws, size_t ws_size,
                              hipStream_t stream) {
  const float* x      = (const float*)d_in[0];
  const int*   neigh  = (const int*)  d_in[1];
  const int*   nbrc   = (const int*)  d_in[2];
  const float* gcnW   = (const float*)d_in[3];   // [3,128,128]
  const float* gcnB   = (const float*)d_in[4];   // [3,128]
  const float* gcnG   = (const float*)d_in[5];
  const float* gcnBb  = (const float*)d_in[6];
  const float* poolW  = (const float*)d_in[7];   // [128,128]
  const float* poolB  = (const float*)d_in[8];
  const float* mrgW   = (const float*)d_in[9];   // [3,128,256]
  const float* mrgB   = (const float*)d_in[10];  // [3,128]
  const float* sgG    = (const float*)d_in[11];  // [2,128]
  const float* sgB    = (const float*)d_in[12];
  float* out = (float*)d_out;                    // [8192, 256]

  const size_t ND = (size_t)NNODE * DIM;
  float*    dinv  = (float*)d_ws;
  unsigned* keep  = (unsigned*)(dinv + NNODE);
  float*    first = (float*)(keep + NNODE);
  float*    nxt   = first + ND;
  float*    hbuf  = nxt + ND;
  float*    sbuf  = hbuf + ND;
  float*    tmp   = sbuf + ND;
  float*    cat   = tmp + ND;                    // [N, 256]

  const dim3 gB(128);                            // 4 waves/block
  const dim3 gG(NNODE / 16, DIM / 64);           // (512, 2) tiles
  const dim3 nB(256);                            // 8 waves = 8 nodes/block
  const dim3 nG(NNODE / 8);

  deg_kernel<<<NNODE / 256, 256, 0, stream>>>(neigh, nbrc, dinv, keep);

  // -------- GCN branch --------
  wmma_gemm_bias<<<gG, gB, 0, stream>>>(x, DIM, gcnW, DIM, gcnB,
                                        first, DIM, DIM, 0);
  gcn_prop_ln<<<nG, nB, 0, stream>>>(first, first, neigh, dinv, keep,
                                     gcnG, gcnBb, hbuf, DIM);

  wmma_gemm_bias<<<gG, gB, 0, stream>>>(hbuf, DIM, gcnW + DIM * DIM, DIM,
                                        gcnB + DIM, nxt, DIM, DIM, 0);
  gcn_prop_ln<<<nG, nB, 0, stream>>>(nxt, hbuf, neigh, dinv, keep,
                                     gcnG + DIM, gcnBb + DIM, hbuf, DIM);

  wmma_gemm_bias<<<gG, gB, 0, stream>>>(hbuf, DIM, gcnW + 2 * DIM * DIM, DIM,
                                        gcnB + 2 * DIM, nxt, DIM, DIM, 0);
  gcn_prop_ln<<<nG, nB, 0, stream>>>(nxt, hbuf, neigh, dinv, keep,
                                     gcnG + 2 * DIM, gcnBb + 2 * DIM,
                                     out, 2 * DIM);                // out[:,0:128]

  // -------- SAGE branch --------
  wmma_gemm_bias<<<gG, gB, 0, stream>>>(x, DIM, poolW, DIM, poolB,
                                        tmp, DIM, DIM, 1);         // ne1
  sage_pool_cat<<<nG, nB, 0, stream>>>(x, tmp, neigh, nbrc, cat);
  wmma_gemm_bias<<<gG, gB, 0, stream>>>(cat, 2 * DIM, mrgW, 2 * DIM, mrgB,
                                        sbuf, DIM, 2 * DIM, 1);    // s1

  // layer 2
  sage_pool_cat<<<nG, nB, 0, stream>>>(sbuf, sbuf, neigh, nbrc, cat);
  wmma_gemm_bias<<<gG, gB, 0, stream>>>(cat, 2 * DIM, mrgW + DIM * 2 * DIM,
                                        2 * DIM, mrgB + DIM, tmp, DIM,
                                        2 * DIM, 1);
  resid_ln<<<nG, nB, 0, stream>>>(tmp, sbuf, sgG, sgB, sbuf, DIM);

  // layer 3
  sage_pool_cat<<<nG, nB, 0, stream>>>(sbuf, sbuf, neigh, nbrc, cat);
  wmma_gemm_bias<<<gG, gB, 0, stream>>>(cat, 2 * DIM, mrgW + 2 * DIM * 2 * DIM,
                                        2 * DIM, mrgB + 2 * DIM, tmp, DIM,
                                        2 * DIM, 1);
  resid_ln<<<nG, nB, 0, stream>>>(tmp, sbuf, sgG + DIM, sgB + DIM,
                                  out + DIM, 2 * DIM);             // out[:,128:256]
}